// SimpleGAT_70841190580754
// MI455X (gfx1250) — compile-verified
//
#include <hip/hip_runtime.h>
#include <math.h>

// ---------------------------------------------------------------------------
// GAT 2-layer forward for MI455X (gfx1250, wave32).
// GEMMs: native fp32 WMMA (V_WMMA_F32_16X16X4_F32), one wave per 16-row strip.
// Edge softmax/aggregation: L2-resident float atomics; per-edge float4
// vectorization (global_load_b128) to minimize instructions around atomics.
// ---------------------------------------------------------------------------

typedef __attribute__((ext_vector_type(2))) float v2f;
typedef __attribute__((ext_vector_type(8))) float v8f;

#define NEG_SLOPE 0.2f

__device__ __forceinline__ float lrelu(float x) { return x >= 0.f ? x : NEG_SLOPE * x; }

// float atomic max via signed/unsigned integer atomics (valid for finite floats,
// destination initialized to -inf).
__device__ __forceinline__ void atomicMaxF(float* addr, float val) {
  if (val >= 0.f) {
    atomicMax((int*)addr, __float_as_int(val));
  } else {
    atomicMin((unsigned int*)addr, __float_as_uint(val));
  }
}

__device__ __forceinline__ void edge_sd(const long long* ei, int E, int e,
                                        int& s, int& d) {
  if (e < E) { s = (int)ei[e]; d = (int)ei[E + e]; }
  else       { s = e - E;      d = e - E; }
}

// ---------------------------------------------------------------------------
// init: zero accumulators, set segment-max buffers to -inf (float4 stores)
// ---------------------------------------------------------------------------
__global__ void k_init(float4* out1, float4* out2, float4* m1, float4* s1,
                       float* m2, float* s2, int N) {
  int i = blockIdx.x * blockDim.x + threadIdx.x;
  const float4 z4 = make_float4(0.f, 0.f, 0.f, 0.f);
  const float ninf = -__builtin_inff();
  if (i < N * 64) out1[i] = z4;                       // N*256 floats
  if (i < N * 10) out2[i] = z4;                       // N*40  floats
  if (i < N) {                                        // N*4 floats each
    m1[i] = make_float4(ninf, ninf, ninf, ninf);
    s1[i] = z4;
    m2[i] = ninf;
    s2[i] = 0.f;
  }
}

// ---------------------------------------------------------------------------
// GEMM1: H[N,256] = X[N,256] @ W[256,256]  (fp32 WMMA 16x16x4)
// One wave per block; blockIdx -> (rowTile, colGroup of 64 cols).
// Fragment layout per CDNA5 ISA 7.12.2.
// ---------------------------------------------------------------------------
__global__ void k_gemm1(const float* __restrict__ X, const float* __restrict__ W,
                        float* __restrict__ H) {
  const int K = 256, NC = 256;
  int rowTile  = blockIdx.x >> 2;
  int colGroup = blockIdx.x & 3;
  int lane = threadIdx.x;
  int half = lane >> 4;
  int l    = lane & 15;
  int row  = rowTile * 16 + l;

  v8f acc0 = {}, acc1 = {}, acc2 = {}, acc3 = {};
  v8f* accs[4] = {&acc0, &acc1, &acc2, &acc3};

  for (int k0 = 0; k0 < K; k0 += 4) {
    int ka = k0 + 2 * half;
    v2f a;
    a.x = X[row * K + ka];
    a.y = X[row * K + ka + 1];
#pragma unroll
    for (int j = 0; j < 4; ++j) {
      int col = colGroup * 64 + j * 16 + l;
      v2f b;
      b.x = W[ka * NC + col];
      b.y = W[(ka + 1) * NC + col];
      *accs[j] = __builtin_amdgcn_wmma_f32_16x16x4_f32(
          false, a, false, b, (short)0, *accs[j], false, false);
    }
  }
#pragma unroll
  for (int j = 0; j < 4; ++j) {
#pragma unroll
    for (int i = 0; i < 8; ++i) {
      int r = rowTile * 16 + i + 8 * half;
      int c = colGroup * 64 + j * 16 + l;
      H[r * NC + c] = (*accs[j])[i];
    }
  }
}

// ---------------------------------------------------------------------------
// GEMM2: H[N,40] = X[N,256] @ W[256,40]  (cols padded to 48, masked)
// ---------------------------------------------------------------------------
__global__ void k_gemm2(const float* __restrict__ X, const float* __restrict__ W,
                        float* __restrict__ H) {
  const int K = 256, NC = 40;
  int rowTile = blockIdx.x;
  int lane = threadIdx.x;
  int half = lane >> 4;
  int l    = lane & 15;
  int row  = rowTile * 16 + l;

  v8f acc0 = {}, acc1 = {}, acc2 = {};
  v8f* accs[3] = {&acc0, &acc1, &acc2};

  for (int k0 = 0; k0 < K; k0 += 4) {
    int ka = k0 + 2 * half;
    v2f a;
    a.x = X[row * K + ka];
    a.y = X[row * K + ka + 1];
#pragma unroll
    for (int j = 0; j < 3; ++j) {
      int col = j * 16 + l;
      v2f b;
      b.x = (col < NC) ? W[ka * NC + col] : 0.f;
      b.y = (col < NC) ? W[(ka + 1) * NC + col] : 0.f;
      *accs[j] = __builtin_amdgcn_wmma_f32_16x16x4_f32(
          false, a, false, b, (short)0, *accs[j], false, false);
    }
  }
#pragma unroll
  for (int j = 0; j < 3; ++j) {
#pragma unroll
    for (int i = 0; i < 8; ++i) {
      int r = rowTile * 16 + i + 8 * half;
      int c = j * 16 + l;
      if (c < NC) H[r * NC + c] = (*accs[j])[i];
    }
  }
}

// ---------------------------------------------------------------------------
// Attention coefficients
// ---------------------------------------------------------------------------
__global__ void k_alpha1(const float* __restrict__ h1, const float* __restrict__ aS,
                         const float* __restrict__ aD, float* as1, float* ad1, int N) {
  int idx = blockIdx.x * blockDim.x + threadIdx.x;
  if (idx >= N * 4) return;
  int n = idx >> 2, h = idx & 3;
  const float* hv = h1 + (size_t)n * 256 + h * 64;
  const float* sv = aS + h * 64;
  const float* dv = aD + h * 64;
  float ss = 0.f, sd = 0.f;
#pragma unroll 8
  for (int d = 0; d < 64; ++d) { float v = hv[d]; ss += v * sv[d]; sd += v * dv[d]; }
  as1[idx] = ss;
  ad1[idx] = sd;
}

__global__ void k_alpha2(const float* __restrict__ h2, const float* __restrict__ aS,
                         const float* __restrict__ aD, float* as2, float* ad2, int N) {
  int n = blockIdx.x * blockDim.x + threadIdx.x;
  if (n >= N) return;
  const float* hv = h2 + (size_t)n * 40;
  float ss = 0.f, sd = 0.f;
#pragma unroll 8
  for (int d = 0; d < 40; ++d) { float v = hv[d]; ss += v * aS[d]; sd += v * aD[d]; }
  as2[n] = ss;
  ad2[n] = sd;
}

// ---------------------------------------------------------------------------
// Layer 1 edge passes (1 thread per edge, all 4 heads via float4)
// Pass A: compute e = lrelu(as[src]+ad[dst]), cache into ex1, segment max.
// Pass B: read cached e, exponentiate, segment sum, overwrite ex1 with exp.
// ---------------------------------------------------------------------------
__global__ void k_edge_max1(const long long* __restrict__ ei, int E, int ET,
                            const float4* __restrict__ as1, const float4* __restrict__ ad1,
                            float* __restrict__ m1, float4* __restrict__ ex1) {
  int e = blockIdx.x * blockDim.x + threadIdx.x;
  if (e >= ET) return;
  int s, d; edge_sd(ei, E, e, s, d);
  float4 a = as1[s], b = ad1[d];
  float4 ev;
  ev.x = lrelu(a.x + b.x);
  ev.y = lrelu(a.y + b.y);
  ev.z = lrelu(a.z + b.z);
  ev.w = lrelu(a.w + b.w);
  ex1[e] = ev;                      // cache raw logits (coalesced)
  float* md = m1 + (size_t)d * 4;
  atomicMaxF(md + 0, ev.x);
  atomicMaxF(md + 1, ev.y);
  atomicMaxF(md + 2, ev.z);
  atomicMaxF(md + 3, ev.w);
}

__global__ void k_edge_exp1(const long long* __restrict__ ei, int E, int ET,
                            const float4* __restrict__ m1, float* __restrict__ s1,
                            float4* __restrict__ ex1) {
  int e = blockIdx.x * blockDim.x + threadIdx.x;
  if (e >= ET) return;
  int d = (e < E) ? (int)ei[E + e] : (e - E);
  float4 ev = ex1[e];
  float4 mv = m1[d];
  float4 ex;
  ex.x = __expf(ev.x - mv.x);
  ex.y = __expf(ev.y - mv.y);
  ex.z = __expf(ev.z - mv.z);
  ex.w = __expf(ev.w - mv.w);
  ex1[e] = ex;
  float* sd = s1 + (size_t)d * 4;
  atomicAdd(sd + 0, ex.x);
  atomicAdd(sd + 1, ex.y);
  atomicAdd(sd + 2, ex.z);
  atomicAdd(sd + 3, ex.w);
}

// One edge per 64-thread block; each thread scatters 4 channels (b128 gather).
__global__ void k_scatter1(const long long* __restrict__ ei, int E,
                           const float4* __restrict__ h1, const float4* __restrict__ ex1,
                           const float4* __restrict__ s1, float* __restrict__ out1) {
  int e = blockIdx.x;
  __shared__ int ss, sd;
  __shared__ float alpha[4];
  if (threadIdx.x == 0) {
    int a, b; edge_sd(ei, E, e, a, b);
    ss = a; sd = b;
  }
  __syncthreads();
  if (threadIdx.x == 0) {
    float4 ex = ex1[e];
    float4 sv = s1[sd];
    alpha[0] = ex.x / sv.x;
    alpha[1] = ex.y / sv.y;
    alpha[2] = ex.z / sv.z;
    alpha[3] = ex.w / sv.w;
  }
  __syncthreads();
  int t = threadIdx.x;              // 0..63 -> channels 4t..4t+3
  int head = t >> 4;                // 16 threads per 64-channel head
  float a = alpha[head];
  float4 hv = h1[(size_t)ss * 64 + t];
  float* ob = out1 + (size_t)sd * 256 + t * 4;
  atomicAdd(ob + 0, hv.x * a);
  atomicAdd(ob + 1, hv.y * a);
  atomicAdd(ob + 2, hv.z * a);
  atomicAdd(ob + 3, hv.w * a);
}

// out1 = relu(out1 + bias1)   (layer-2 input, in place, float4)
__global__ void k_relu1(float4* out1, const float4* __restrict__ bias1, int N) {
  int i = blockIdx.x * blockDim.x + threadIdx.x;
  if (i >= N * 64) return;
  float4 v = out1[i];
  float4 b = bias1[i & 63];
  v.x = fmaxf(v.x + b.x, 0.f);
  v.y = fmaxf(v.y + b.y, 0.f);
  v.z = fmaxf(v.z + b.z, 0.f);
  v.w = fmaxf(v.w + b.w, 0.f);
  out1[i] = v;
}

// ---------------------------------------------------------------------------
// Layer 2 edge passes (1 head, 40 dims)
// ---------------------------------------------------------------------------
__global__ void k_edge_max2(const long long* __restrict__ ei, int E, int ET,
                            const float* __restrict__ as2, const float* __restrict__ ad2,
                            float* __restrict__ m2, float* __restrict__ ex2) {
  int e = blockIdx.x * blockDim.x + threadIdx.x;
  if (e >= ET) return;
  int s, d; edge_sd(ei, E, e, s, d);
  float ev = lrelu(as2[s] + ad2[d]);
  ex2[e] = ev;                      // cache raw logit
  atomicMaxF(&m2[d], ev);
}

__global__ void k_edge_exp2(const long long* __restrict__ ei, int E, int ET,
                            const float* __restrict__ m2, float* __restrict__ s2,
                            float* __restrict__ ex2) {
  int e = blockIdx.x * blockDim.x + threadIdx.x;
  if (e >= ET) return;
  int d = (e < E) ? (int)ei[E + e] : (e - E);
  float ex = __expf(ex2[e] - m2[d]);
  ex2[e] = ex;
  atomicAdd(&s2[d], ex);
}

// Thread per (edge, quad-of-4-channels): 10 quads cover the 40 dims.
__global__ void k_scatter2(const long long* __restrict__ ei, int E, int ET,
                           const float4* __restrict__ h2, const float* __restrict__ ex2,
                           const float* __restrict__ s2, float* __restrict__ out2) {
  int idx = blockIdx.x * blockDim.x + threadIdx.x;
  if (idx >= ET * 10) return;
  int e = idx / 10, q = idx - e * 10;
  int s, d; edge_sd(ei, E, e, s, d);
  float a = ex2[e] / s2[d];
  float4 hv = h2[(size_t)s * 10 + q];
  float* ob = out2 + (size_t)d * 40 + q * 4;
  atomicAdd(ob + 0, hv.x * a);
  atomicAdd(ob + 1, hv.y * a);
  atomicAdd(ob + 2, hv.z * a);
  atomicAdd(ob + 3, hv.w * a);
}

// bias + log_softmax over 40 classes -> final output
__global__ void k_final(const float* __restrict__ out2, const float* __restrict__ bias2,
                        float* __restrict__ y, int N) {
  int n = blockIdx.x * blockDim.x + threadIdx.x;
  if (n >= N) return;
  float v[40];
  float m = -__builtin_inff();
#pragma unroll
  for (int c = 0; c < 40; ++c) { v[c] = out2[(size_t)n * 40 + c] + bias2[c]; m = fmaxf(m, v[c]); }
  float s = 0.f;
#pragma unroll
  for (int c = 0; c < 40; ++c) s += __expf(v[c] - m);
  float ls = m + __logf(s);
#pragma unroll
  for (int c = 0; c < 40; ++c) y[(size_t)n * 40 + c] = v[c] - ls;
}

// ---------------------------------------------------------------------------
// Host launcher
// ---------------------------------------------------------------------------
extern "C" void kernel_launch(void* const* d_in, const int* in_sizes, int n_in,
                              void* d_out, int out_size, void* d_ws, size_t ws_size,
                              hipStream_t stream) {
  const float*     x    = (const float*)d_in[0];
  const long long* ei   = (const long long*)d_in[1];   // [2, E] int64
  const float*     W1   = (const float*)d_in[2];       // [256, 256]
  const float*     aS1  = (const float*)d_in[3];       // [4, 64]
  const float*     aD1  = (const float*)d_in[4];       // [4, 64]
  const float*     b1   = (const float*)d_in[5];       // [256]
  const float*     W2   = (const float*)d_in[6];       // [256, 40]
  const float*     aS2  = (const float*)d_in[7];       // [40]
  const float*     aD2  = (const float*)d_in[8];       // [40]
  const float*     b2   = (const float*)d_in[9];       // [40]

  const int IN = 256, HID = 256, OUT = 40;
  const int N  = in_sizes[0] / IN;       // 10000 (divisible by 16)
  const int E  = in_sizes[1] / 2;        // 320000
  const int ET = E + N;                  // + self loops

  // workspace carve-up (floats; every buffer 16B-aligned by construction)
  float* p = (float*)d_ws;
  float* h1   = p; p += (size_t)N * HID;
  float* as1  = p; p += (size_t)N * 4;
  float* ad1  = p; p += (size_t)N * 4;
  float* m1   = p; p += (size_t)N * 4;
  float* s1   = p; p += (size_t)N * 4;
  float* ex1  = p; p += (size_t)ET * 4;
  float* out1 = p; p += (size_t)N * HID;
  float* h2   = p; p += (size_t)N * OUT;
  float* as2  = p; p += (size_t)N;
  float* ad2  = p; p += (size_t)N;
  float* m2   = p; p += (size_t)N;
  float* s2   = p; p += (size_t)N;
  float* ex2  = p; p += (size_t)ET;
  float* out2 = p; p += (size_t)N * OUT;
  float* y    = (float*)d_out;

  const int T = 256;
  k_init<<<(N * 64 + T - 1) / T, T, 0, stream>>>((float4*)out1, (float4*)out2,
                                                 (float4*)m1, (float4*)s1, m2, s2, N);

  // ---- layer 1 ----
  k_gemm1<<<(N / 16) * 4, 32, 0, stream>>>(x, W1, h1);
  k_alpha1<<<(N * 4 + T - 1) / T, T, 0, stream>>>(h1, aS1, aD1, as1, ad1, N);
  k_edge_max1<<<(ET + T - 1) / T, T, 0, stream>>>(ei, E, ET, (const float4*)as1,
                                                  (const float4*)ad1, m1, (float4*)ex1);
  k_edge_exp1<<<(ET + T - 1) / T, T, 0, stream>>>(ei, E, ET, (const float4*)m1,
                                                  s1, (float4*)ex1);
  k_scatter1<<<ET, 64, 0, stream>>>(ei, E, (const float4*)h1, (const float4*)ex1,
                                    (const float4*)s1, out1);
  k_relu1<<<(N * 64 + T - 1) / T, T, 0, stream>>>((float4*)out1, (const float4*)b1, N);

  // ---- layer 2 ----
  k_gemm2<<<N / 16, 32, 0, stream>>>(out1, W2, h2);
  k_alpha2<<<(N + T - 1) / T, T, 0, stream>>>(h2, aS2, aD2, as2, ad2, N);
  k_edge_max2<<<(ET + T - 1) / T, T, 0, stream>>>(ei, E, ET, as2, ad2, m2, ex2);
  k_edge_exp2<<<(ET + T - 1) / T, T, 0, stream>>>(ei, E, ET, m2, s2, ex2);
  k_scatter2<<<(ET * 10 + T - 1) / T, T, 0, stream>>>(ei, E, ET, (const float4*)h2,
                                                      ex2, s2, out2);
  k_final<<<(N + T - 1) / T, T, 0, stream>>>(out2, b2, y, N);
}